// DisparityWarp_21114059227575
// MI455X (gfx1250) — compile-verified
//
#include <hip/hip_runtime.h>

// DisparityWarp: out[b,c,h,w] = bilinear sample of src[b,c] at
//   ix = (w - disp[b,0,h,w]) * W/(W-1) - 0.5,  iy = h * H/(H-1) - 0.5
// zero padding outside. Memory-bound: one thread per pixel, weights shared
// across C=32 channels, NT stores for the streaming output.

namespace {
constexpr int Bn = 8;
constexpr int Cn = 32;
constexpr int Hn = 384;
constexpr int Wn = 768;
constexpr int HW = Hn * Wn;          // 294912
constexpr int CHW = Cn * HW;         // per-batch plane stride
constexpr int NPIX = Bn * Hn * Wn;   // 2,359,296 (== 9216 * 256)
}

__global__ __launch_bounds__(256) void disparity_warp_kernel(
    const float* __restrict__ src,
    const float* __restrict__ disp,
    float* __restrict__ out)
{
    const int pix = blockIdx.x * 256 + threadIdx.x;   // flat (b,h,w)
    if (pix >= NPIX) return;

    const int w  = pix % Wn;          // W=768 = 3*256 -> block stays in one row
    const int bh = pix / Wn;
    const int h  = bh % Hn;
    const int b  = bh / Hn;

    // disparity layout [B,1,H,W] flattens to exactly `pix`; single use -> NT load
    const float d = __builtin_nontemporal_load(disp + pix);

    // unnormalized sample coords (align_corners=False algebra from reference)
    const float ix = ((float)w - d) * (float(Wn) / float(Wn - 1)) - 0.5f;
    const float iy = (float)h       * (float(Hn) / float(Hn - 1)) - 0.5f;

    const float x0f = floorf(ix);
    const float y0f = floorf(iy);
    const float fx = ix - x0f;
    const float fy = iy - y0f;
    const int x0 = (int)x0f;
    const int y0 = (int)y0f;
    const int x1 = x0 + 1;
    const int y1 = y0 + 1;

    // zero-padding masks folded into the bilinear weights
    const float mx0 = (x0 >= 0 && x0 < Wn) ? 1.0f : 0.0f;
    const float mx1 = (x1 >= 0 && x1 < Wn) ? 1.0f : 0.0f;
    const float my0 = (y0 >= 0 && y0 < Hn) ? 1.0f : 0.0f;
    const float my1 = (y1 >= 0 && y1 < Hn) ? 1.0f : 0.0f;

    const float gx0 = 1.0f - fx;
    const float gy0 = 1.0f - fy;
    const float w00 = gy0 * gx0 * (my0 * mx0);
    const float w01 = gy0 * fx  * (my0 * mx1);
    const float w10 = fy  * gx0 * (my1 * mx0);
    const float w11 = fy  * fx  * (my1 * mx1);

    // clamped corner offsets within one channel plane
    const int cx0 = min(max(x0, 0), Wn - 1);
    const int cx1 = min(max(x1, 0), Wn - 1);
    const int cy0 = min(max(y0, 0), Hn - 1);
    const int cy1 = min(max(y1, 0), Hn - 1);
    const int o00 = cy0 * Wn + cx0;
    const int o01 = cy0 * Wn + cx1;
    const int o10 = cy1 * Wn + cx0;
    const int o11 = cy1 * Wn + cx1;

    const float* __restrict__ pb = src + (size_t)b * CHW;
    float* __restrict__ qb = out + (size_t)b * CHW + (size_t)h * Wn + w;

    // amortize weights over all 32 channels; unroll x8 for MLP (32 loads in
    // flight per thread) without blowing VGPR budget under wave32.
#pragma unroll 8
    for (int c = 0; c < Cn; ++c) {
        const float* __restrict__ pc = pb + (size_t)c * HW;
        const float v = w00 * pc[o00] + w01 * pc[o01] +
                        w10 * pc[o10] + w11 * pc[o11];
        // output is write-once streaming: non-temporal store keeps src resident
        __builtin_nontemporal_store(v, qb + (size_t)c * HW);
    }
}

extern "C" void kernel_launch(void* const* d_in, const int* in_sizes, int n_in,
                              void* d_out, int out_size, void* d_ws, size_t ws_size,
                              hipStream_t stream)
{
    (void)in_sizes; (void)n_in; (void)out_size; (void)d_ws; (void)ws_size;
    const float* src  = (const float*)d_in[0];
    const float* disp = (const float*)d_in[1];
    float* out = (float*)d_out;

    const int threads = 256;
    const int blocks  = (NPIX + threads - 1) / threads;   // 9216
    disparity_warp_kernel<<<blocks, threads, 0, stream>>>(src, disp, out);
}